// MLP1_14010183320171
// MI455X (gfx1250) — compile-verified
//
#include <hip/hip_runtime.h>

typedef __attribute__((ext_vector_type(2))) float v2f;
typedef __attribute__((ext_vector_type(8))) float v8f;

#define BATCH   8
#define CIN     6
#define PADC    8                      // feat channels padded to 8 (WMMA K = 2x4)
#define NPTS    4096
#define KNN     10
#define OC      64
#define TOTCOL  (BATCH * NPTS * KNN)   // 327680 columns of the big GEMM
#define NEG_SLOPE 0.2f
#define EPSBN   1e-5f

// ---------------------------------------------------------------------------
// K1: KNN. One block = 256 query points of one batch. Batch xyz staged in LDS.
// pd = 2*<q,p> - |q|^2 - |p|^2  (negative squared distance); keep top-10.
// ---------------------------------------------------------------------------
__global__ __launch_bounds__(256) void knn_kernel(const float* __restrict__ x,
                                                  int* __restrict__ idx_out) {
    __shared__ float sx[NPTS * 3];
    const int b = blockIdx.y;
    const float* xb = x + (size_t)b * CIN * NPTS;
    for (int i = threadIdx.x; i < 3 * NPTS; i += 256) {
        int c = i / NPTS, m = i - c * NPTS;
        sx[m * 3 + c] = xb[c * NPTS + m];
    }
    __syncthreads();

    const int n = blockIdx.x * 256 + threadIdx.x;
    const float qx = sx[n * 3 + 0], qy = sx[n * 3 + 1], qz = sx[n * 3 + 2];
    const float qq = qx * qx + qy * qy + qz * qz;

    float bval[KNN];
    int   bidx[KNN];
#pragma unroll
    for (int j = 0; j < KNN; ++j) { bval[j] = -3.402823466e38f; bidx[j] = 0; }

    for (int m = 0; m < NPTS; ++m) {
        float px = sx[m * 3 + 0], py = sx[m * 3 + 1], pz = sx[m * 3 + 2];
        float inner = qx * px + qy * py + qz * pz;
        float pp = px * px + py * py + pz * pz;
        float pd = 2.0f * inner - qq - pp;
        if (pd > bval[0]) {           // bval sorted ascending; bval[0] is current min
            bool placed = false;
#pragma unroll
            for (int j = 0; j < KNN; ++j) {
                if (!placed) {
                    if (j < KNN - 1 && pd > bval[j + 1]) {
                        bval[j] = bval[j + 1]; bidx[j] = bidx[j + 1];
                    } else {
                        bval[j] = pd; bidx[j] = m; placed = true;
                    }
                }
            }
        }
    }
    int* op = idx_out + (size_t)(b * NPTS + n) * KNN;
#pragma unroll
    for (int j = 0; j < KNN; ++j) op[j] = bidx[KNN - 1 - j];  // descending, like top_k
}

// ---------------------------------------------------------------------------
// K2: gather neighbors, center xyz over k and scale by 10, write feat[c][col],
// col = (b*N + n)*K + k  (channel-major so GEMM B-tiles are contiguous).
// Channels 6,7 are written as zeros (WMMA K-padding) -> branch-free B loads.
// ---------------------------------------------------------------------------
__global__ __launch_bounds__(256) void gather_kernel(const float* __restrict__ x,
                                                     const int* __restrict__ idx,
                                                     float* __restrict__ feat) {
    int t = blockIdx.x * 256 + threadIdx.x;
    if (t >= BATCH * NPTS) return;
    int b = t >> 12;
    const float* xb = x + (size_t)b * CIN * NPTS;
    const int* ip = idx + (size_t)t * KNN;

    int id[KNN];
#pragma unroll
    for (int j = 0; j < KNN; ++j) id[j] = ip[j];

    float g[CIN][KNN];
#pragma unroll
    for (int c = 0; c < CIN; ++c)
#pragma unroll
        for (int j = 0; j < KNN; ++j) g[c][j] = xb[c * NPTS + id[j]];

#pragma unroll
    for (int c = 0; c < 3; ++c) {
        float s = 0.0f;
#pragma unroll
        for (int j = 0; j < KNN; ++j) s += g[c][j];
        float mn = s * (1.0f / KNN);
#pragma unroll
        for (int j = 0; j < KNN; ++j) g[c][j] = (g[c][j] - mn) * 10.0f;
    }
    size_t colb = (size_t)t * KNN;
#pragma unroll
    for (int c = 0; c < CIN; ++c)
#pragma unroll
        for (int j = 0; j < KNN; ++j) feat[(size_t)c * TOTCOL + colb + j] = g[c][j];
    // zero K-padding channels 6,7
#pragma unroll
    for (int c = CIN; c < PADC; ++c)
#pragma unroll
        for (int j = 0; j < KNN; ++j) feat[(size_t)c * TOTCOL + colb + j] = 0.0f;
}

// ---------------------------------------------------------------------------
// K3a: partial first/second moments of feat over all columns (fp64 accum).
// 27 quantities: s1[6], upper-tri s2[21]. One partial row per block.
// ---------------------------------------------------------------------------
__global__ __launch_bounds__(256) void moments_kernel(const float* __restrict__ feat,
                                                      double* __restrict__ partial) {
    double s1[6], s2[21];
#pragma unroll
    for (int q = 0; q < 6; ++q) s1[q] = 0.0;
#pragma unroll
    for (int q = 0; q < 21; ++q) s2[q] = 0.0;

    const int stride = gridDim.x * 256;
    for (int col = blockIdx.x * 256 + threadIdx.x; col < TOTCOL; col += stride) {
        float f[6];
#pragma unroll
        for (int c = 0; c < 6; ++c) f[c] = feat[(size_t)c * TOTCOL + col];
#pragma unroll
        for (int c = 0; c < 6; ++c) s1[c] += (double)f[c];
        int q = 0;
#pragma unroll
        for (int c = 0; c < 6; ++c)
#pragma unroll
            for (int c2 = c; c2 < 6; ++c2) s2[q++] += (double)f[c] * (double)f[c2];
    }

    __shared__ double red[8][27];
    const int lane = threadIdx.x & 31, wave = threadIdx.x >> 5;
#pragma unroll
    for (int q = 0; q < 27; ++q) {
        double v = (q < 6) ? s1[q] : s2[q - 6];
        for (int off = 16; off > 0; off >>= 1) v += __shfl_down(v, off, 32);
        if (lane == 0) red[wave][q] = v;
    }
    __syncthreads();
    if (threadIdx.x < 27) {
        double s = 0.0;
        for (int w = 0; w < 8; ++w) s += red[w][threadIdx.x];
        partial[(size_t)blockIdx.x * 27 + threadIdx.x] = s;
    }
}

// ---------------------------------------------------------------------------
// K3b: finalize BN stats. mu_o = W_o . E[feat], E[y^2]_o = W_o^T M2 W_o.
// stats[0..63] = mu, stats[64..127] = rsqrt(var + eps).
// ---------------------------------------------------------------------------
__global__ __launch_bounds__(64) void stats_kernel(const double* __restrict__ partial, int nb,
                                                   const float* __restrict__ W,
                                                   float* __restrict__ stats) {
    __shared__ double tot[27];
    if (threadIdx.x < 27) {
        double s = 0.0;
        for (int p = 0; p < nb; ++p) s += partial[(size_t)p * 27 + threadIdx.x];
        tot[threadIdx.x] = s;
    }
    __syncthreads();
    const int o = threadIdx.x;
    if (o < OC) {
        float w[6];
#pragma unroll
        for (int c = 0; c < 6; ++c) w[c] = W[o * 6 + c];
        const double invT = 1.0 / (double)TOTCOL;
        double mu = 0.0;
#pragma unroll
        for (int c = 0; c < 6; ++c) mu += (double)w[c] * tot[c];
        mu *= invT;
        double ey2 = 0.0;
        int q = 6;
#pragma unroll
        for (int c = 0; c < 6; ++c)
#pragma unroll
            for (int c2 = c; c2 < 6; ++c2) {
                double coeff = (c == c2) ? 1.0 : 2.0;
                ey2 += coeff * (double)w[c] * (double)w[c2] * tot[q];
                ++q;
            }
        ey2 *= invT;
        double var = ey2 - mu * mu;
        stats[o] = (float)mu;
        stats[OC + o] = rsqrtf((float)var + EPSBN);
    }
}

// ---------------------------------------------------------------------------
// K4: y = W @ feat via V_WMMA_F32_16X16X4_F32, fused BN + leaky-ReLU + max over k.
// Block = 64 threads (2 waves); each wave owns 8 consecutive n (80 columns =
// 5 col-tiles), 4 o-tiles of 16, 2 K-chunks of 4 (channels padded to 8).
// A: lane l, comp j -> W[16*ot + l%16, kc*4 + 2*(l/16) + j]
// B: lane l, comp j -> feat[kc*4 + 2*(l/16) + j, coltile + l%16]
// D: vgpr r, lane l -> y[16*ot + r + 8*(l/16), coltile + l%16]
// ---------------------------------------------------------------------------
__global__ __launch_bounds__(64) void gemm_bn_max_kernel(const float* __restrict__ feat,
                                                         const float* __restrict__ W,
                                                         const float* __restrict__ stats,
                                                         const float* __restrict__ gamma,
                                                         const float* __restrict__ beta,
                                                         float* __restrict__ z) {
    __shared__ float sW[OC * 8];
    __shared__ float sy[2][OC * 80];
    for (int i = threadIdx.x; i < OC * 8; i += 64) {
        int o = i >> 3, c = i & 7;
        sW[i] = (c < CIN) ? W[o * CIN + c] : 0.0f;
    }
    __syncthreads();

    const int wave = threadIdx.x >> 5, lane = threadIdx.x & 31;
    const int half = lane >> 4, lm = lane & 15;
    const int b = blockIdx.y;
    const int n0 = blockIdx.x * 16 + wave * 8;
    const size_t colbase = (size_t)(b * NPTS + n0) * KNN;   // 80 contiguous columns
    float* yw = &sy[wave][0];

    v2f A[4][2];
#pragma unroll
    for (int ot = 0; ot < 4; ++ot)
#pragma unroll
        for (int kc = 0; kc < 2; ++kc) {
            int row = ot * 16 + lm;
            int c0 = kc * 4 + half * 2;
            A[ot][kc].x = sW[row * 8 + c0];
            A[ot][kc].y = sW[row * 8 + c0 + 1];
        }

    for (int ct = 0; ct < 5; ++ct) {
        const size_t col = colbase + ct * 16 + lm;
        v2f Bv[2];
#pragma unroll
        for (int kc = 0; kc < 2; ++kc) {
            int c0 = kc * 4 + half * 2;           // 0,2,4,6 -- all valid (padded feat)
            Bv[kc].x = feat[(size_t)c0 * TOTCOL + col];
            Bv[kc].y = feat[(size_t)(c0 + 1) * TOTCOL + col];
        }
#pragma unroll
        for (int ot = 0; ot < 4; ++ot) {
            v8f acc = {};
            acc = __builtin_amdgcn_wmma_f32_16x16x4_f32(false, A[ot][0], false, Bv[0],
                                                        (short)0, acc, false, false);
            acc = __builtin_amdgcn_wmma_f32_16x16x4_f32(false, A[ot][1], false, Bv[1],
                                                        (short)0, acc, false, false);
#pragma unroll
            for (int r = 0; r < 8; ++r) {
                int o = ot * 16 + half * 8 + r;
                float v = (acc[r] - stats[o]) * stats[OC + o] * gamma[o] + beta[o];
                v = (v >= 0.0f) ? v : NEG_SLOPE * v;
                yw[o * 80 + ct * 16 + lm] = v;
            }
        }
    }
    __syncthreads();

    // max over k for this wave's 8 n's, all 64 channels
    for (int e = lane; e < OC * 8; e += 32) {
        int o = e >> 3, nl = e & 7;
        const float* p = &yw[o * 80 + nl * KNN];
        float m = p[0];
#pragma unroll
        for (int k = 1; k < KNN; ++k) m = fmaxf(m, p[k]);
        z[(size_t)(b * OC + o) * NPTS + (n0 + nl)] = m;
    }
}

// ---------------------------------------------------------------------------
// K5: per (b,o): max and mean over n (deterministic tree reduction).
// out[b*128 + o] = max, out[b*128 + 64 + o] = mean.
// ---------------------------------------------------------------------------
__global__ __launch_bounds__(256) void out_reduce_kernel(const float* __restrict__ z,
                                                         float* __restrict__ out) {
    const int bo = blockIdx.x;
    const int b = bo >> 6, o = bo & 63;
    const float* zp = z + (size_t)bo * NPTS;
    float mx = -3.402823466e38f, sm = 0.0f;
    for (int n = threadIdx.x; n < NPTS; n += 256) {
        float v = zp[n];
        mx = fmaxf(mx, v);
        sm += v;
    }
    __shared__ float smx[256], ssm[256];
    smx[threadIdx.x] = mx;
    ssm[threadIdx.x] = sm;
    __syncthreads();
    for (int s = 128; s > 0; s >>= 1) {
        if (threadIdx.x < s) {
            smx[threadIdx.x] = fmaxf(smx[threadIdx.x], smx[threadIdx.x + s]);
            ssm[threadIdx.x] += ssm[threadIdx.x + s];
        }
        __syncthreads();
    }
    if (threadIdx.x == 0) {
        out[(size_t)b * 128 + o] = smx[0];
        out[(size_t)b * 128 + 64 + o] = ssm[0] * (1.0f / NPTS);
    }
}

// ---------------------------------------------------------------------------
extern "C" void kernel_launch(void* const* d_in, const int* in_sizes, int n_in,
                              void* d_out, int out_size, void* d_ws, size_t ws_size,
                              hipStream_t stream) {
    const float* x     = (const float*)d_in[0];   // (8,6,4096)
    const float* W     = (const float*)d_in[1];   // (64,6)
    const float* gamma = (const float*)d_in[2];   // (64,)
    const float* beta  = (const float*)d_in[3];   // (64,)
    float* out = (float*)d_out;                   // (8,128)

    // workspace layout (~20.3 MB total, all 8-byte aligned)
    char* ws = (char*)d_ws;
    size_t off = 0;
    int*    d_idx   = (int*)(ws + off);    off += (size_t)TOTCOL * sizeof(int);           // 1.25 MB
    float*  d_feat  = (float*)(ws + off);  off += (size_t)PADC * TOTCOL * sizeof(float);  // 10.5 MB
    double* d_part  = (double*)(ws + off); off += (size_t)512 * 27 * sizeof(double);      // 108 KB
    float*  d_stats = (float*)(ws + off);  off += 2 * OC * sizeof(float) + 256;           // pad
    float*  d_z     = (float*)(ws + off);  // 8*64*4096 floats = 8.4 MB

    knn_kernel<<<dim3(NPTS / 256, BATCH), 256, 0, stream>>>(x, d_idx);
    gather_kernel<<<(BATCH * NPTS) / 256, 256, 0, stream>>>(x, d_idx, d_feat);
    moments_kernel<<<512, 256, 0, stream>>>(d_feat, d_part);
    stats_kernel<<<1, 64, 0, stream>>>(d_part, 512, W, d_stats);
    gemm_bn_max_kernel<<<dim3(NPTS / 16, BATCH), 64, 0, stream>>>(d_feat, W, d_stats,
                                                                  gamma, beta, d_z);
    out_reduce_kernel<<<BATCH * OC, 256, 0, stream>>>(d_z, out);
}